// KDECapsuleLayer_6743098655304
// MI455X (gfx1250) — compile-verified
//
#include <hip/hip_runtime.h>
#include <math.h>

// B=64, N=128, IN_DIM=768, D=64
typedef float v2f __attribute__((ext_vector_type(2)));
typedef float v8f __attribute__((ext_vector_type(8)));

__device__ __forceinline__ float wave_sum(float x) {
  x += __shfl_xor(x, 16);
  x += __shfl_xor(x, 8);
  x += __shfl_xor(x, 4);
  x += __shfl_xor(x, 2);
  x += __shfl_xor(x, 1);
  return x;
}
__device__ __forceinline__ float wave_max(float x) {
  x = fmaxf(x, __shfl_xor(x, 16));
  x = fmaxf(x, __shfl_xor(x, 8));
  x = fmaxf(x, __shfl_xor(x, 4));
  x = fmaxf(x, __shfl_xor(x, 2));
  x = fmaxf(x, __shfl_xor(x, 1));
  return x;
}

// LDS flat address -> LDS byte offset (low 32 bits of the flat address)
__device__ __forceinline__ unsigned lds_off(const void* p) {
  return (unsigned)(size_t)p;
}
// async 16B global->LDS copy, GVS mode: saddr(SGPR64) + vaddr(VGPR32 byte offset)
__device__ __forceinline__ void async_copy16(unsigned lds_addr, unsigned goff,
                                             const void* sbase) {
  asm volatile("global_load_async_to_lds_b128 %0, %1, %2"
               :: "v"(lds_addr), "v"(goff), "s"(sbase) : "memory");
}
__device__ __forceinline__ void async_wait0() {
  asm volatile("s_wait_asynccnt 0x0" ::: "memory");
}

// ---------------- Attention GEMM1: partial[gk] = flat[64,768slice] @ w1_sliceT ----
__global__ void k_attn1(const float* __restrict__ x, const float* __restrict__ w1,
                        float* __restrict__ partial) {
  __shared__ __align__(16) float As[64 * 17];    // [m][k], pitch 17 (bank-conflict pad)
  __shared__ __align__(16) float Bs[128 * 17];   // [n][k], pitch 17
  const int gk = blockIdx.x;              // 128 K-slices of 768 (K=98304)
  const int t = threadIdx.x;              // 256 threads, 8 waves
  const int w = t >> 5, l = t & 31;
  const int mt = w & 3, ntb = (w >> 2) * 4;
  const int lm = l & 15, lh = l >> 4;
  const unsigned asB = lds_off(As), bsB = lds_off(Bs);
  v8f acc[4] = {};
  for (int ch = 0; ch < 48; ++ch) {
    const int k0 = gk * 768 + ch * 16;
    {
      int m = t >> 2, kq = t & 3;         // 64x16 A tile: 256 chunks of 16B
      async_copy16(asB + (unsigned)(m * 17 + kq * 4) * 4u,
                   (unsigned)(m * 98304 + k0 + kq * 4) * 4u, x);
    }
#pragma unroll
    for (int i = 0; i < 2; ++i) {         // 128x16 W tile: 512 chunks of 16B
      int idx = t + i * 256;
      int n = idx >> 2, kq = idx & 3;
      async_copy16(bsB + (unsigned)(n * 17 + kq * 4) * 4u,
                   (unsigned)(n * 98304 + k0 + kq * 4) * 4u, w1);
    }
    async_wait0();
    __syncthreads();
#pragma unroll
    for (int kk = 0; kk < 16; kk += 4) {
      v2f a;
      a.x = As[(16 * mt + lm) * 17 + kk + 2 * lh + 0];
      a.y = As[(16 * mt + lm) * 17 + kk + 2 * lh + 1];
#pragma unroll
      for (int q = 0; q < 4; ++q) {
        v2f bf;   // B[k][n] = w1[n][k] read transposed out of row-major Bs
        bf.x = Bs[(16 * (ntb + q) + lm) * 17 + kk + 2 * lh + 0];
        bf.y = Bs[(16 * (ntb + q) + lm) * 17 + kk + 2 * lh + 1];
        acc[q] = __builtin_amdgcn_wmma_f32_16x16x4_f32(false, a, false, bf,
                                                       (short)0, acc[q], false, false);
      }
    }
    __syncthreads();
  }
#pragma unroll
  for (int q = 0; q < 4; ++q)
#pragma unroll
    for (int r = 0; r < 8; ++r) {
      int row = 16 * mt + r + 8 * lh;     // 0..63
      int col = 16 * (ntb + q) + lm;      // 0..127
      partial[gk * 8192 + row * 128 + col] = acc[q][r];
    }
}

// Deterministic fixed-order reduction of K-slice partials + bias + relu
__global__ void k_attn1red(const float* __restrict__ partial, const float* __restrict__ b1,
                           float* __restrict__ att1) {
  int i = blockIdx.x * 256 + threadIdx.x;   // 8192
  float s = 0.f;
  for (int g = 0; g < 128; ++g) s += partial[g * 8192 + i];
  s += b1[i & 127];
  att1[i] = fmaxf(s, 0.0f);
}

__global__ void k_attn2(const float* __restrict__ att1, const float* __restrict__ w2,
                        const float* __restrict__ b2, float* __restrict__ att) {
  int idx = blockIdx.x * 256 + threadIdx.x; // 8192
  int b = idx >> 7, n = idx & 127;
  const float* ar = att1 + b * 128;
  const float* wr = w2 + n * 128;
  float s = 0.f;
  for (int k = 0; k < 128; ++k) s += ar[k] * wr[k];
  att[idx] = tanhf(s + b2[n]);
}

// xs = x*(1+att), norms = ||row||
__global__ void k_scale(const float* __restrict__ x, const float* __restrict__ att,
                        float* __restrict__ xs, float* __restrict__ norms) {
  int w = threadIdx.x >> 5, l = threadIdx.x & 31;
  int row = blockIdx.x * 8 + w;             // 8192 (b,n) rows
  float a = 1.0f + att[row];
  const float* xr = x + (size_t)row * 768;
  float* xo = xs + (size_t)row * 768;
  float s = 0.f;
  for (int k = l; k < 768; k += 32) {
    float v = xr[k] * a;
    xo[k] = v;
    s += v * v;
  }
  s = wave_sum(s);
  if (l == 0) norms[row] = sqrtf(s);
}

// top-64-of-128 mask (ties broken by lower index, matching lax.top_k)
__global__ void k_topk(const float* __restrict__ norms, float* __restrict__ mask) {
  __shared__ float sn[128];
  int b = blockIdx.x, n = threadIdx.x;
  sn[n] = norms[b * 128 + n];
  __syncthreads();
  float me = sn[n];
  int rank = 0;
  for (int j = 0; j < 128; ++j) {
    float o = sn[j];
    rank += (o > me) || (o == me && j < n);
  }
  mask[b * 128 + n] = (rank < 64) ? 1.0f : 0.0f;
}

// ------------- Prediction GEMM: p = (xs @ W^T + bias) * mask -> d_out p region ----
__global__ void k_pred(const float* __restrict__ A, const float* __restrict__ Wm,
                       const float* __restrict__ bias, const float* __restrict__ mask,
                       float* __restrict__ out) {
  __shared__ __align__(16) float As[128 * 17];   // [m][k], pitch 17
  __shared__ __align__(16) float Bs[128 * 17];   // [n][k], pitch 17
  const int m0 = blockIdx.x * 128;          // rows (b*N+i), 8192 total
  const int n0 = blockIdx.y * 128;          // cols (j*D+d), 8192 total
  const int t = threadIdx.x;
  const int w = t >> 5, l = t & 31;
  const int lm = l & 15, lh = l >> 4;
  const unsigned asB = lds_off(As), bsB = lds_off(Bs);
  v8f acc[8] = {};
  for (int ch = 0; ch < 48; ++ch) {
    const int k0 = ch * 16;
#pragma unroll
    for (int i = 0; i < 2; ++i) {           // 2x 512 chunks of 16B
      int idx = t + i * 256;
      int m = idx >> 2, kq = idx & 3;
      async_copy16(asB + (unsigned)(m * 17 + kq * 4) * 4u,
                   (unsigned)((m0 + m) * 768 + k0 + kq * 4) * 4u, A);
      async_copy16(bsB + (unsigned)(m * 17 + kq * 4) * 4u,
                   (unsigned)((n0 + m) * 768 + k0 + kq * 4) * 4u, Wm);
    }
    if (ch + 1 < 48) {  // global_prefetch next K-chunk
      __builtin_prefetch(A + (size_t)(m0 + (t >> 1)) * 768 + k0 + 16, 0, 0);
      __builtin_prefetch(Wm + (size_t)(n0 + (t >> 1)) * 768 + k0 + 16, 0, 0);
    }
    async_wait0();
    __syncthreads();
#pragma unroll
    for (int kk = 0; kk < 16; kk += 4) {
      v2f a;
      a.x = As[(16 * w + lm) * 17 + kk + 2 * lh + 0];
      a.y = As[(16 * w + lm) * 17 + kk + 2 * lh + 1];
#pragma unroll
      for (int nt = 0; nt < 8; ++nt) {
        v2f bf;
        bf.x = Bs[(16 * nt + lm) * 17 + kk + 2 * lh + 0];
        bf.y = Bs[(16 * nt + lm) * 17 + kk + 2 * lh + 1];
        acc[nt] = __builtin_amdgcn_wmma_f32_16x16x4_f32(false, a, false, bf,
                                                        (short)0, acc[nt], false, false);
      }
    }
    __syncthreads();
  }
#pragma unroll
  for (int nt = 0; nt < 8; ++nt)
#pragma unroll
    for (int r = 0; r < 8; ++r) {
      int row = m0 + 16 * w + r + 8 * lh;
      int col = n0 + 16 * nt + lm;
      float val = (acc[nt][r] + bias[(row & 127) * 8192 + col]) * mask[row];
      out[(size_t)row * 8192 + col] = val;
    }
}

// ---------------- Routing precompute: ||u||^2 and squash scale per (b,i,j) --------
__global__ void k_prep(const float* __restrict__ u, const float* __restrict__ mask,
                       float* __restrict__ nu2, float* __restrict__ su) {
  int w = threadIdx.x >> 5, l = threadIdx.x & 31;
  int row = blockIdx.x * 8 + w;             // 1048576 rows (b,i,j)
  float mk = mask[row >> 7];                // (b*N+i)
  float n2 = 0.f;
  if (mk != 0.f) {
    const float* ur = u + (size_t)row * 64;
    float u0 = ur[l], u1 = ur[l + 32];
    n2 = wave_sum(u0 * u0 + u1 * u1);
  }
  if (l == 0) {
    nu2[row] = n2;
    su[row] = (n2 > 0.f) ? (sqrtf(n2) / (0.5f + n2)) : 0.f;
  }
}

// r1: softmax over j (128 real cols + implicit trailing 0 logit)
__global__ void r1_softmax(const float* __restrict__ scal, const float* __restrict__ coef,
                           float* __restrict__ c) {
  if (scal[0] != 0.f) return;
  int w = threadIdx.x >> 5, l = threadIdx.x & 31;
  int row = blockIdx.x * 8 + w;             // 8192 (b,i)
  const float* cr = coef + row * 128;
  float v0 = cr[l], v1 = cr[l + 32], v2 = cr[l + 64], v3 = cr[l + 96];
  float m = fmaxf(fmaxf(v0, v1), fmaxf(v2, v3));
  m = wave_max(fmaxf(m, 0.0f));
  float e0 = expf(v0 - m), e1 = expf(v1 - m), e2 = expf(v2 - m), e3 = expf(v3 - m);
  float tot = wave_sum(e0 + e1 + e2 + e3) + expf(-m);   // implicit 0 logit
  float* co = c + row * 128;
  co[l] = e0 / tot; co[l + 32] = e1 / tot; co[l + 64] = e2 / tot; co[l + 96] = e3 / tot;
}

// r2: v[b,j,:] = squash( sum_i c[b,i,j]*u[b,i,j,:] )   (masked i skipped)
__global__ void r2_agg(const float* __restrict__ scal, const float* __restrict__ u,
                       const float* __restrict__ c, const float* __restrict__ mask,
                       float* __restrict__ v) {
  if (scal[0] != 0.f) return;
  int w = threadIdx.x >> 5, l = threadIdx.x & 31;
  int pair = blockIdx.x * 8 + w;            // 8192 (b,j)
  int b = pair >> 7, j = pair & 127;
  float s0 = 0.f, s1 = 0.f;
  for (int i = 0; i < 128; ++i) {
    int bi = b * 128 + i;
    if (mask[bi] != 0.f) {
      float cij = c[bi * 128 + j];
      const float* ur = u + ((size_t)bi * 128 + j) * 64;
      s0 += cij * ur[l];
      s1 += cij * ur[l + 32];
    }
  }
  float n2 = wave_sum(s0 * s0 + s1 * s1);
  float sc = sqrtf(n2) / (0.5f + n2);
  float* vr = v + ((size_t)b * 128 + j) * 64;
  vr[l] = s0 * sc;
  vr[l + 32] = s1 * sc;
}

// r3: d = 1 - ||su*u - v||, coef += d, per-(b,i) kde partial (fixed order)
__global__ void r3_dist(const float* __restrict__ scal, const float* __restrict__ u,
                        const float* __restrict__ v, const float* __restrict__ mask,
                        const float* __restrict__ nu2, const float* __restrict__ su,
                        float* __restrict__ coef, float* __restrict__ kde_part) {
  if (scal[0] != 0.f) return;
  __shared__ __align__(16) float vt[8192];
  __shared__ float vn2[128];
  __shared__ float wpart[8];
  int t = threadIdx.x;
  int bi = blockIdx.x;                      // 8192 (b,i)
  int b = bi >> 7;
  const float* vb = v + (size_t)b * 8192;
  {                                         // async-stage the first 8KB (j < 32) of v[b]
    const unsigned vtB = lds_off(vt);
#pragma unroll
    for (int i = 0; i < 2; ++i) {
      int idx = t + i * 256;                // 512 chunks of 16B
      async_copy16(vtB + (unsigned)idx * 16u, (unsigned)idx * 16u, vb);
    }
    async_wait0();
  }
  __syncthreads();
  if (t < 128) {
    float s = 0.f;
    const float* vj = vt + t * 64;
    for (int d = 0; d < 64; ++d) s += vj[d] * vj[d];
    vn2[t] = s;
  }
  __syncthreads();
  int w = t >> 5, l = t & 31;
  float mk = mask[bi];
  float kacc = 0.f;
  for (int jj = 0; jj < 16; ++jj) {
    int j = w * 16 + jj;
    float dot = 0.f, suv = 0.f, n2v = 0.f;
    if (mk != 0.f) {
      const float* ur = u + ((size_t)bi * 128 + j) * 64;
      float p = ur[l] * vt[j * 64 + l] + ur[l + 32] * vt[j * 64 + l + 32];
      dot = wave_sum(p);
      suv = su[bi * 128 + j];
      n2v = nu2[bi * 128 + j];
    }
    float t2 = suv * suv * n2v - 2.f * suv * dot + vn2[j];
    float dval = 1.f - sqrtf(fmaxf(t2, 0.f));
    if (l == 0) {
      int ci = bi * 128 + j;
      float co = coef[ci];
      kacc += (co + dval) * dval;
      coef[ci] = co + dval;
    }
  }
  if (l == 0) wpart[w] = kacc;
  __syncthreads();
  if (t == 0) {
    float s = 0.f;
    for (int q = 0; q < 8; ++q) s += wpart[q];
    kde_part[bi] = s;
  }
}

// r4: deterministic global KDE reduction + convergence flag update
__global__ void r4_kde(const float* __restrict__ kde_part, float* __restrict__ scal) {
  if (scal[0] != 0.f) return;
  __shared__ float red[256];
  int t = threadIdx.x;
  float s = 0.f;
  for (int g = 0; g < 32; ++g) s += kde_part[t + g * 256];
  red[t] = s;
  __syncthreads();
  for (int st = 128; st > 0; st >>= 1) {
    if (t < st) red[t] += red[t + st];
    __syncthreads();
  }
  if (t == 0) {
    float kde = logf(red[0] * (1.0f / 64.0f));
    float last = scal[1];
    if (fabsf(kde - last) < 0.05f) scal[0] = 1.0f;
    scal[1] = kde;
  }
}

__global__ void k_act(const float* __restrict__ v, float* __restrict__ out) {
  int idx = blockIdx.x * 256 + threadIdx.x; // 8192 (b,j)
  const float* vr = v + (size_t)idx * 64;
  float s = 0.f;
  for (int d = 0; d < 64; ++d) s += vr[d] * vr[d];
  float a = sqrtf(s);
  out[idx] = fminf(fmaxf(a, 1e-6f), 1.0f - 1e-6f);
}

extern "C" void kernel_launch(void* const* d_in, const int* in_sizes, int n_in,
                              void* d_out, int out_size, void* d_ws, size_t ws_size,
                              hipStream_t stream) {
  const float* x    = (const float*)d_in[0];
  const float* w1   = (const float*)d_in[1];
  const float* b1   = (const float*)d_in[2];
  const float* w2   = (const float*)d_in[3];
  const float* b2   = (const float*)d_in[4];
  const float* W    = (const float*)d_in[5];
  const float* bias = (const float*)d_in[6];
  float* out_act = (float*)d_out;
  float* out_p   = (float*)d_out + 8192;    // u = p lives in the output buffer

  float* ws      = (float*)d_ws;
  float* partial = ws;                       // 1048576  (reused as nu2 later)
  float* nu2p    = ws;
  float* att1    = ws + 1048576;             // 8192
  float* att     = att1 + 8192;              // 8192
  float* norms   = att + 8192;               // 8192
  float* maskp   = norms + 8192;             // 8192
  float* xs      = maskp + 8192;             // 6291456 (dead after k_pred; reused:)
  float* sup     = xs;                       //   1048576
  float* cp      = sup + 1048576;            //   1048576
  float* coef    = cp + 1048576;             //   1048576
  float* scal    = coef + 1048576;           //   4  (done, last)
  float* vp      = scal + 4;                 //   524288
  float* kdep    = vp + 524288;              //   8192

  k_attn1   <<<128,  256, 0, stream>>>(x, w1, partial);
  k_attn1red<<<32,   256, 0, stream>>>(partial, b1, att1);
  k_attn2   <<<32,   256, 0, stream>>>(att1, w2, b2, att);
  k_scale   <<<1024, 256, 0, stream>>>(x, att, xs, norms);
  k_topk    <<<64,   128, 0, stream>>>(norms, maskp);
  k_pred    <<<dim3(64, 64), 256, 0, stream>>>(xs, W, bias, maskp, out_p);
  // xs region now dead -> init routing state (coef + scalars)
  hipMemsetAsync(coef, 0, (1048576 + 4) * sizeof(float), stream);
  k_prep    <<<131072, 256, 0, stream>>>(out_p, maskp, nu2p, sup);
  for (int it = 0; it < 12; ++it) {
    r1_softmax<<<1024, 256, 0, stream>>>(scal, coef, cp);
    r2_agg    <<<1024, 256, 0, stream>>>(scal, out_p, cp, maskp, vp);
    r3_dist   <<<8192, 256, 0, stream>>>(scal, out_p, vp, maskp, nu2p, sup, coef, kdep);
    r4_kde    <<<1,    256, 0, stream>>>(kdep, scal);
  }
  k_act<<<32, 256, 0, stream>>>(vp, out_act);
}